// PointTransformerLayer_59115929862838
// MI455X (gfx1250) — compile-verified
//
#include <hip/hip_runtime.h>
#include <hip/hip_bf16.h>
#include <math.h>

// ---------------------------------------------------------------------------
// Point Transformer layer for MI455X (gfx1250), fp32 everywhere.
// Heavy math (34 GFLOP attn MLP + projections) runs on V_WMMA_F32_16X16X4_F32,
// with double-buffered GLOBAL_LOAD_ASYNC_TO_LDS staging (ASYNCcnt pipeline).
// ---------------------------------------------------------------------------

typedef __attribute__((ext_vector_type(2))) float v2f;
typedef __attribute__((ext_vector_type(8))) float v8f;
typedef __attribute__((ext_vector_type(4))) int   v4i;

constexpr int NB    = 4;       // batch
constexpr int NPTS  = 4096;    // points per batch
constexpr int DIMC  = 128;     // channels
constexpr int KNN   = 16;      // neighbors
constexpr int PH    = 32;      // pos mlp hidden
constexpr int AH    = 256;     // attn mlp hidden
constexpr int PTOT  = NB * NPTS;        // 16384
constexpr int RTOT  = PTOT * KNN;       // 262144

// Direct global->LDS async copy (no VGPR round-trip, tracked by ASYNCcnt).
__device__ __forceinline__ void async_cp16(void* lds, const void* g) {
    __builtin_amdgcn_global_load_async_to_lds_b128(
        (__attribute__((address_space(1))) v4i*)(g),
        (__attribute__((address_space(3))) v4i*)(lds),
        0, 0);
}

// ------------------------------- kNN ---------------------------------------
// One thread per query point; candidates tiled through LDS; register
// insertion-sorted top-16 (strict < keeps lax.top_k tie order; self included).
__global__ __launch_bounds__(128) void knn_kernel(const float* __restrict__ pos,
                                                  int* __restrict__ knn_idx) {
    __shared__ float sx[128], sy[128], sz[128];
    const int b = blockIdx.y;
    const int n = blockIdx.x * 128 + threadIdx.x;
    const float* pb = pos + (size_t)b * NPTS * 3;
    const float px = pb[n * 3 + 0];
    const float py = pb[n * 3 + 1];
    const float pz = pb[n * 3 + 2];

    float bd[KNN];
    int   bi[KNN];
#pragma unroll
    for (int t = 0; t < KNN; ++t) { bd[t] = 3.4e38f; bi[t] = 0; }

    for (int jt = 0; jt < NPTS; jt += 128) {
        const int j = jt + threadIdx.x;
        sx[threadIdx.x] = pb[j * 3 + 0];
        sy[threadIdx.x] = pb[j * 3 + 1];
        sz[threadIdx.x] = pb[j * 3 + 2];
        __syncthreads();
        for (int jj = 0; jj < 128; ++jj) {
            const float dx = px - sx[jj];
            const float dy = py - sy[jj];
            const float dz = pz - sz[jj];
            const float d  = dx * dx + dy * dy + dz * dz;
            if (d < bd[KNN - 1]) {
                float cd = d; int ci = jt + jj;
#pragma unroll
                for (int t = 0; t < KNN; ++t) {
                    if (cd < bd[t]) {
                        float td = bd[t]; int ti = bi[t];
                        bd[t] = cd; bi[t] = ci;
                        cd = td; ci = ti;
                    }
                }
            }
        }
        __syncthreads();
    }
    int* op = knn_idx + ((size_t)b * NPTS + n) * KNN;
#pragma unroll
    for (int t = 0; t < KNN; ++t) op[t] = bi[t];
}

// ------------------------- fp32 WMMA GEMM ----------------------------------
// out[M,N] = act(A[M,K] @ W[K,N] + bias[N]);  M%128==0, N%64==0, K%16==0.
// Block 256 threads = 8 waves; block tile 128x64; wave tile 16x64 (4 accum
// tiles).  K staged in 16-wide slabs via double-buffered async-to-LDS copies:
// prefetch slab k+1 while WMMAs consume slab k; s_wait_asynccnt 3 leaves only
// the in-flight prefetch outstanding.
constexpr int BM = 128;
constexpr int BN = 64;
constexpr int BK = 16;
constexpr int APAD = 20;  // sA row stride (floats), padded vs 16
constexpr int WPAD = 68;  // sW row stride (floats), padded vs 64

__global__ __launch_bounds__(256) void gemm_wmma_f32(
    const float* __restrict__ A, const float* __restrict__ W,
    const float* __restrict__ bias, float* __restrict__ out,
    int M, int N, int K, int do_relu) {
    __shared__ float sA[2][BM * APAD];
    __shared__ float sW[2][BK * WPAD];

    const int tid  = threadIdx.x;
    const int wave = tid >> 5;
    const int lane = tid & 31;
    const int m0   = blockIdx.x * BM;
    const int n0   = blockIdx.y * BN;
    const int wm   = wave * 16;          // wave's M-subtile inside block
    const int lh   = lane >> 4;          // lane half (0/1)
    const int ll   = lane & 15;

    // staging coordinates (3 async b128 copies per thread per slab)
    const int ar = tid >> 2;             // 0..63  (A rows, 2 rounds)
    const int ac = (tid & 3) * 4;        // A col group
    const int wr = tid >> 4;             // 0..15  (W row)
    const int wc = (tid & 15) * 4;       // W col group

    v8f acc[4] = {};

    auto stage = [&](int kb, int buf) {
#pragma unroll
        for (int rr = 0; rr < 2; ++rr) {
            const int row = ar + rr * 64;
            async_cp16(&sA[buf][row * APAD + ac],
                       A + (size_t)(m0 + row) * K + kb + ac);
        }
        async_cp16(&sW[buf][wr * WPAD + wc],
                   W + (size_t)(kb + wr) * N + n0 + wc);
    };

    stage(0, 0);
    const int nk = K / BK;
    for (int ks = 0; ks < nk; ++ks) {
        const int buf = ks & 1;
        if (ks + 1 < nk) {
            stage((ks + 1) * BK, buf ^ 1);
            __builtin_amdgcn_s_wait_asynccnt(3);   // slab ks landed; prefetch in flight
        } else {
            __builtin_amdgcn_s_wait_asynccnt(0);
        }
        __syncthreads();

        const float* cA = sA[buf];
        const float* cW = sW[buf];
#pragma unroll
        for (int kk = 0; kk < BK; kk += 4) {
            // A fragment (16x4): lanes0-15 -> K{kk,kk+1}, lanes16-31 -> K{kk+2,kk+3}
            const int acol = kk + 2 * lh;              // even -> 8B aligned
            const v2f afrag = *(const v2f*)(cA + (wm + ll) * APAD + acol);
            const int brow = kk + 2 * lh;
#pragma unroll
            for (int t = 0; t < 4; ++t) {
                const int bcol = t * 16 + ll;
                v2f bfrag;
                bfrag.x = cW[brow * WPAD + bcol];
                bfrag.y = cW[(brow + 1) * WPAD + bcol];
                acc[t] = __builtin_amdgcn_wmma_f32_16x16x4_f32(
                    false, afrag, false, bfrag, (short)0, acc[t], false, false);
            }
        }
        __syncthreads();
    }

    // epilogue: C/D layout -> row = wm + r + 8*lanehalf, col = t*16 + (lane&15)
#pragma unroll
    for (int t = 0; t < 4; ++t) {
        const int col = n0 + t * 16 + ll;
        const float bv = bias ? bias[col] : 0.0f;
#pragma unroll
        for (int r = 0; r < 8; ++r) {
            const int row = m0 + wm + r + 8 * lh;
            float v = acc[t][r] + bv;
            if (do_relu) v = fmaxf(v, 0.0f);
            out[(size_t)row * N + col] = v;
        }
    }
}

// ---------------- pos MLP layer 1: rel(3) -> relu(32) -----------------------
// 32 threads per (b,n,k) row; thread = one hidden unit (3 FMA).
__global__ __launch_bounds__(256) void penc1_kernel(
    const float* __restrict__ pos, const int* __restrict__ knn_idx,
    const float* __restrict__ w1, const float* __restrict__ b1,
    float* __restrict__ p1) {
    const int gid = blockIdx.x * 256 + threadIdx.x;
    const int row = gid >> 5;          // (b,n,k) flat
    const int hu  = gid & 31;
    const int pnt = row >> 4;          // b*NPTS + n
    const int b   = pnt / NPTS;
    const int n   = pnt % NPTS;
    const int j   = knn_idx[row];
    const float* pb = pos + (size_t)b * NPTS * 3;
    const float rx = pb[n * 3 + 0] - pb[j * 3 + 0];
    const float ry = pb[n * 3 + 1] - pb[j * 3 + 1];
    const float rz = pb[n * 3 + 2] - pb[j * 3 + 2];
    const float a = rx * w1[hu] + ry * w1[32 + hu] + rz * w1[64 + hu] + b1[hu];
    p1[(size_t)row * PH + hu] = fmaxf(a, 0.0f);
}

// ---------------- h = q - gather(k) + pos_enc -------------------------------
__global__ __launch_bounds__(256) void hbuild_kernel(
    const float* __restrict__ q, const float* __restrict__ kp,
    const float* __restrict__ pe, const int* __restrict__ knn_idx,
    float* __restrict__ h) {
    const size_t e = (size_t)blockIdx.x * 256 + threadIdx.x;  // RTOT*128 elems
    const int row = (int)(e >> 7);
    const int d   = (int)(e & 127);
    const int pnt = row >> 4;
    const int b   = pnt / NPTS;
    const int j   = knn_idx[row];
    h[e] = q[(size_t)pnt * DIMC + d]
         - kp[((size_t)b * NPTS + j) * DIMC + d]
         + pe[e];
}

// ------------- softmax over K (per channel) + aggregate ---------------------
__global__ __launch_bounds__(128) void smax_kernel(
    const float* __restrict__ logits, const float* __restrict__ pe,
    const float* __restrict__ vp, const int* __restrict__ knn_idx,
    float* __restrict__ out) {
    const int pnt = blockIdx.x;            // b*NPTS + n
    const int d   = threadIdx.x;
    const int b   = pnt / NPTS;
    const size_t base = (size_t)pnt * KNN;
    const float scale = 0.08838834764831845f;  // 1/sqrt(128)

    float l[KNN];
    float m = -3.4e38f;
#pragma unroll
    for (int k = 0; k < KNN; ++k) {
        l[k] = logits[(base + k) * DIMC + d] * scale;
        m = fmaxf(m, l[k]);
    }
    float s = 0.0f;
#pragma unroll
    for (int k = 0; k < KNN; ++k) { l[k] = expf(l[k] - m); s += l[k]; }
    const float inv = 1.0f / s;
    float acc = 0.0f;
#pragma unroll
    for (int k = 0; k < KNN; ++k) {
        const int j = knn_idx[base + k];
        acc += l[k] * inv * (vp[((size_t)b * NPTS + j) * DIMC + d] +
                             pe[(base + k) * DIMC + d]);
    }
    out[(size_t)pnt * DIMC + d] = acc;
}

// ---------------------------------------------------------------------------
extern "C" void kernel_launch(void* const* d_in, const int* in_sizes, int n_in,
                              void* d_out, int out_size, void* d_ws, size_t ws_size,
                              hipStream_t stream) {
    const float* x    = (const float*)d_in[0];
    const float* pos  = (const float*)d_in[1];
    const float* Wq   = (const float*)d_in[2];
    const float* Wk   = (const float*)d_in[3];
    const float* Wv   = (const float*)d_in[4];
    const float* p_w1 = (const float*)d_in[5];
    const float* p_b1 = (const float*)d_in[6];
    const float* p_w2 = (const float*)d_in[7];
    const float* p_b2 = (const float*)d_in[8];
    const float* a_w1 = (const float*)d_in[9];
    const float* a_b1 = (const float*)d_in[10];
    const float* a_w2 = (const float*)d_in[11];
    const float* a_b2 = (const float*)d_in[12];
    float* out = (float*)d_out;

    char* ws = (char*)d_ws;
    size_t off = 0;
    auto take = [&](size_t bytes) {
        char* p = ws + off;
        off = (off + bytes + 255) & ~(size_t)255;
        return p;
    };
    int*   idx = (int*)  take((size_t)RTOT * sizeof(int));
    float* q   = (float*)take((size_t)PTOT * DIMC * sizeof(float));
    float* kp  = (float*)take((size_t)PTOT * DIMC * sizeof(float));
    float* vp  = (float*)take((size_t)PTOT * DIMC * sizeof(float));
    float* p1  = (float*)take((size_t)RTOT * PH   * sizeof(float));
    float* pe  = (float*)take((size_t)RTOT * DIMC * sizeof(float));
    float* h   = (float*)take((size_t)RTOT * DIMC * sizeof(float));  // reused for logits
    float* hid = (float*)take((size_t)RTOT * AH   * sizeof(float));
    (void)ws_size; (void)n_in; (void)in_sizes; (void)out_size;

    // 1) brute-force kNN
    knn_kernel<<<dim3(NPTS / 128, NB), 128, 0, stream>>>(pos, idx);

    // 2) q/k/v projections (WMMA)
    gemm_wmma_f32<<<dim3(PTOT / BM, DIMC / BN), 256, 0, stream>>>(x, Wq, nullptr, q,  PTOT, DIMC, DIMC, 0);
    gemm_wmma_f32<<<dim3(PTOT / BM, DIMC / BN), 256, 0, stream>>>(x, Wk, nullptr, kp, PTOT, DIMC, DIMC, 0);
    gemm_wmma_f32<<<dim3(PTOT / BM, DIMC / BN), 256, 0, stream>>>(x, Wv, nullptr, vp, PTOT, DIMC, DIMC, 0);

    // 3) pos MLP: layer1 VALU, layer2 WMMA
    penc1_kernel<<<(RTOT * 32) / 256, 256, 0, stream>>>(pos, idx, p_w1, p_b1, p1);
    gemm_wmma_f32<<<dim3(RTOT / BM, DIMC / BN), 256, 0, stream>>>(p1, p_w2, p_b2, pe, RTOT, DIMC, PH, 0);

    // 4) h = q - gather(k) + pos_enc
    hbuild_kernel<<<(RTOT * DIMC) / 256, 256, 0, stream>>>(q, kp, pe, idx, h);

    // 5) attn MLP (the 34-GFLOP hot path, WMMA)
    gemm_wmma_f32<<<dim3(RTOT / BM, AH / BN), 256, 0, stream>>>(h, a_w1, a_b1, hid, RTOT, AH, DIMC, 1);
    gemm_wmma_f32<<<dim3(RTOT / BM, DIMC / BN), 256, 0, stream>>>(hid, a_w2, a_b2, h, RTOT, DIMC, AH, 0);

    // 6) softmax over K + aggregate
    smax_kernel<<<PTOT, 128, 0, stream>>>(h, pe, vp, idx, out);
}